// Decoder_62139586838765
// MI455X (gfx1250) — compile-verified
//
#include <hip/hip_runtime.h>

typedef __attribute__((ext_vector_type(2))) float v2f;
typedef __attribute__((ext_vector_type(8))) float v8f;

#define N_NODES 100000
#define N_EDGES 1600000
#define D_IN    64
#define D_H0    256
#define D_H1    1024
#define N_TILES (N_NODES / 16)   // 6250, exact

// ---------------------------------------------------------------------------
// Kernel 1: tiny FC stack  h0 = relu(W0@e + b0), h1 = relu(W1@h0 + b1)
// Single block of 256 threads; everything fits in LDS.
// ---------------------------------------------------------------------------
__global__ __launch_bounds__(256) void k_small_fc(
    const float* __restrict__ emb,
    const float* __restrict__ W0, const float* __restrict__ b0,
    const float* __restrict__ W1, const float* __restrict__ b1,
    float* __restrict__ h1_out)
{
    __shared__ float e_s[D_IN];
    __shared__ float h0_s[D_H0];
    const int tid = threadIdx.x;

    if (tid < D_IN) e_s[tid] = emb[tid];
    __syncthreads();

    {   // h0: one output per thread
        float s = b0[tid];
        const float* w = W0 + tid * D_IN;
        #pragma unroll 8
        for (int j = 0; j < D_IN; ++j) s = fmaf(w[j], e_s[j], s);
        h0_s[tid] = fmaxf(s, 0.0f);
    }
    __syncthreads();

    for (int o = tid; o < D_H1; o += 256) {   // h1: 4 outputs per thread
        float s = b1[o];
        const float* w = W1 + (long)o * D_H0;
        #pragma unroll 8
        for (int j = 0; j < D_H0; ++j) s = fmaf(w[j], h0_s[j], s);
        h1_out[o] = fmaxf(s, 0.0f);
    }
}

// ---------------------------------------------------------------------------
// Kernel 2: big GEMV  x[n] = relu(W2[n,:]·h1 + b2[n])  via V_WMMA_F32_16X16X4_F32
// One 16-row tile per wave, 2 waves (64 threads) per block -> 3125 blocks, no
// partial waves, EXEC is all-1s at every WMMA.
//  A (16x4 f32): lane m=lane&15 -> row, half=lane>>4 -> K pair {0,1}/{2,3}
//  B (4x16 f32): h1[k..k+3] in column N=0 only (lanes 0 and 16 carry data)
//  D (16x16 f32): column N=0 -> lane 0 holds M=0..7 in acc[0..7], lane 16 M=8..15
// ---------------------------------------------------------------------------
__global__ __launch_bounds__(64) void k_fc2_wmma(
    const float* __restrict__ W2, const float* __restrict__ b2,
    const float* __restrict__ h1, float* __restrict__ x_out)
{
    __shared__ float h_s[D_H1];
    const int tid = threadIdx.x;
    for (int j = tid; j < D_H1; j += 64) h_s[j] = h1[j];
    __syncthreads();

    const int wave = tid >> 5;
    const int lane = tid & 31;
    const int half = lane >> 4;    // which K pair of the 16x4 A slice
    const int m    = lane & 15;    // row within tile
    const long tile = (long)blockIdx.x * 2 + wave;      // 0..6249
    const long row  = tile * 16 + m;
    const float* __restrict__ wrow = W2 + row * (long)D_H1;
    const bool lead = (m == 0);    // lanes 0 / 16 carry the B column

    v8f acc = {};
    #pragma unroll 4
    for (int kk = 0; kk < D_H1; kk += 4) {
        const int col = kk + 2 * half;
        v2f a = *(const v2f*)(wrow + col);          // global_load_b64, per-row stream
        v2f hv = *(const v2f*)(&h_s[col]);          // ds_load_b64 (broadcast)
        v2f b;
        b.x = lead ? hv.x : 0.0f;
        b.y = lead ? hv.y : 0.0f;
        acc = __builtin_amdgcn_wmma_f32_16x16x4_f32(
                  false, a, false, b, (short)0, acc, false, false);
    }

    if (m == 0) {   // lane 0 -> rows tile*16+0..7, lane 16 -> rows tile*16+8..15
        const long base = tile * 16 + half * 8;
        #pragma unroll
        for (int r = 0; r < 8; ++r) {
            const long n = base + r;
            x_out[n] = fmaxf(acc[r] + b2[n], 0.0f);
        }
    }
}

// ---------------------------------------------------------------------------
// GCN graph kernels
// ---------------------------------------------------------------------------
__global__ __launch_bounds__(256) void k_deg_init(float* __restrict__ deg)
{
    int n = blockIdx.x * 256 + threadIdx.x;
    if (n < N_NODES) deg[n] = 1.0f;   // self loop
}

__global__ __launch_bounds__(256) void k_deg_count(
    const long long* __restrict__ dst, float* __restrict__ deg)
{
    int e = blockIdx.x * 256 + threadIdx.x;
    if (e < N_EDGES) atomicAdd(&deg[(int)dst[e]], 1.0f);
}

// dinv[n] = rsqrt(deg); out[n] = gcn_b + gcn_w*x[n]*dinv[n]^2  (self-loop term)
__global__ __launch_bounds__(256) void k_norm_init(
    const float* __restrict__ x, const float* __restrict__ deg,
    const float* __restrict__ gw, const float* __restrict__ gb,
    float* __restrict__ dinv, float* __restrict__ out)
{
    int n = blockIdx.x * 256 + threadIdx.x;
    if (n < N_NODES) {
        float di = rsqrtf(deg[n]);    // deg >= 1 always
        dinv[n] = di;
        out[n] = gb[0] + gw[0] * x[n] * di * di;
    }
}

__global__ __launch_bounds__(256) void k_edge_agg(
    const long long* __restrict__ src, const long long* __restrict__ dst,
    const float* __restrict__ x, const float* __restrict__ dinv,
    const float* __restrict__ gw, float* __restrict__ out)
{
    int e = blockIdx.x * 256 + threadIdx.x;
    if (e < N_EDGES) {
        int s = (int)src[e];
        int d = (int)dst[e];
        atomicAdd(&out[d], gw[0] * x[s] * dinv[s] * dinv[d]);
    }
}

// ---------------------------------------------------------------------------
extern "C" void kernel_launch(void* const* d_in, const int* in_sizes, int n_in,
                              void* d_out, int out_size, void* d_ws, size_t ws_size,
                              hipStream_t stream)
{
    const float*      emb = (const float*)d_in[0];
    const long long*  ei  = (const long long*)d_in[1];   // [2, E] int64
    const float*      W0  = (const float*)d_in[2];
    const float*      b0  = (const float*)d_in[3];
    const float*      W1  = (const float*)d_in[4];
    const float*      b1  = (const float*)d_in[5];
    const float*      W2  = (const float*)d_in[6];
    const float*      b2  = (const float*)d_in[7];
    const float*      gw  = (const float*)d_in[8];
    const float*      gb  = (const float*)d_in[9];
    float* out = (float*)d_out;

    float* ws   = (float*)d_ws;
    float* h1   = ws;                  // 1024
    float* x    = h1  + D_H1;          // 100000
    float* deg  = x   + N_NODES;       // 100000
    float* dinv = deg + N_NODES;       // 100000

    const long long* src = ei;
    const long long* dst = ei + N_EDGES;

    k_small_fc<<<1, 256, 0, stream>>>(emb, W0, b0, W1, b1, h1);
    k_deg_init<<<(N_NODES + 255) / 256, 256, 0, stream>>>(deg);
    k_deg_count<<<(N_EDGES + 255) / 256, 256, 0, stream>>>(dst, deg);
    k_fc2_wmma<<<N_TILES / 2, 64, 0, stream>>>(W2, b2, h1, x);
    k_norm_init<<<(N_NODES + 255) / 256, 256, 0, stream>>>(x, deg, gw, gb, dinv, out);
    k_edge_agg<<<(N_EDGES + 255) / 256, 256, 0, stream>>>(src, dst, x, dinv, gw, out);
}